// FWHT_901943132182
// MI455X (gfx1250) — compile-verified
//
#include <hip/hip_runtime.h>
#include <hip/hip_bf16.h>

// FWHT along n=1024 via H_1024 = H_32 (x) H_32, each 32x32 Hadamard multiply
// done with V_WMMA_F32_16X16X4_F32 (full fp32, matches reference precision).
// Memory-bound: 256MB traffic -> ~11us roofline at 23.3 TB/s.

typedef float v2f __attribute__((ext_vector_type(2)));
typedef float v8f __attribute__((ext_vector_type(8)));

#define N_FWHT 1024
#define D_FULL 256
#define DT     16     // d-columns per workgroup tile
#define LROW   16     // LDS row stride in floats (1024*16*4 = 64KB exactly)

__global__ __launch_bounds__(256) void fwht_wmma_kernel(
    const float* __restrict__ x, float* __restrict__ y) {
  __shared__ float lds[N_FWHT * LROW];  // 64 KB tile: [n][d]

  const int tid  = threadIdx.x;
  const int lane = tid & 31;            // wave32
  const int wid  = tid >> 5;            // 8 waves
  const int b    = blockIdx.y;
  const int d0   = blockIdx.x * DT;

  const float* xb = x + (size_t)b * N_FWHT * D_FULL + d0;
  float*       yb = y + (size_t)b * N_FWHT * D_FULL + d0;

  // ---- Load 1024 x 16 tile into LDS (float4 per thread, 64B row segments) ----
  {
    const int c4 = (tid & 3) * 4;       // 0,4,8,12
    const int r0 = tid >> 2;            // 0..63
#pragma unroll
    for (int it = 0; it < 16; ++it) {
      const int n = r0 + it * 64;
      const float4 v = *(const float4*)(xb + (size_t)n * D_FULL + c4);
      *(float4*)(&lds[n * LROW + c4]) = v;
    }
  }
  __syncthreads();

  // ---- Precompute H32 A-fragments (16x4 f32 tiles), layout per ISA 7.12.2:
  //      lane holds M = lane%16, K = 2*(lane>=16) + vgpr.
  //      H32[m][k] = (-1)^popcount(m&k)
  const int mlo   = lane & 15;
  const int khalf = (lane >> 4) * 2;
  v2f hA[2][8];
#pragma unroll
  for (int t = 0; t < 2; ++t)
#pragma unroll
    for (int c = 0; c < 8; ++c) {
      const int m  = t * 16 + mlo;
      const int k0 = c * 4 + khalf;
      hA[t][c][0] = (__builtin_popcount(m & k0) & 1) ? -1.0f : 1.0f;
      hA[t][c][1] = (__builtin_popcount(m & (k0 + 1)) & 1) ? -1.0f : 1.0f;
    }

  // ---- Stage A: for each 32-row block i: T_i = H32 * X_i  (in place; each
  //      wave owns whole disjoint 32-row blocks, so no cross-wave hazard) ----
#pragma unroll
  for (int it = 0; it < 4; ++it) {
    const int i     = wid + it * 8;     // 0..31
    const int nbase = i * 32;

    // B fragments (4x16 f32): N = lane%16 -> d, K = 2*(lane>=16)+vgpr -> j
    v2f Bf[8];
#pragma unroll
    for (int c = 0; c < 8; ++c) {
      const int k = c * 4 + khalf;
      Bf[c][0] = lds[(nbase + k)     * LROW + mlo];
      Bf[c][1] = lds[(nbase + k + 1) * LROW + mlo];
    }

    v8f acc0 = {}; v8f acc1 = {};
#pragma unroll
    for (int c = 0; c < 8; ++c) {
      acc0 = __builtin_amdgcn_wmma_f32_16x16x4_f32(false, hA[0][c], false, Bf[c],
                                                   (short)0, acc0, false, false);
      acc1 = __builtin_amdgcn_wmma_f32_16x16x4_f32(false, hA[1][c], false, Bf[c],
                                                   (short)0, acc1, false, false);
    }

    // C/D layout: M = vgpr + 8*(lane>=16), N = lane%16
    const int rowoff = (lane >> 4) * 8;
#pragma unroll
    for (int r = 0; r < 8; ++r) {
      lds[(nbase + rowoff + r)      * LROW + mlo] = acc0[r];  // j' = 0..15
      lds[(nbase + 16 + rowoff + r) * LROW + mlo] = acc1[r];  // j' = 16..31
    }
  }
  __syncthreads();

  // ---- Stage B: for each j': y[32*i' + j'] = sum_i H32[i'][i] * T[32*i + j'],
  //      results written straight to global memory ----
#pragma unroll
  for (int it = 0; it < 4; ++it) {
    const int jp = wid + it * 8;        // j' = 0..31

    v2f Bf[8];
#pragma unroll
    for (int c = 0; c < 8; ++c) {
      const int k = c * 4 + khalf;      // k = i (outer index)
      Bf[c][0] = lds[(k       * 32 + jp) * LROW + mlo];
      Bf[c][1] = lds[((k + 1) * 32 + jp) * LROW + mlo];
    }

    v8f acc0 = {}; v8f acc1 = {};
#pragma unroll
    for (int c = 0; c < 8; ++c) {
      acc0 = __builtin_amdgcn_wmma_f32_16x16x4_f32(false, hA[0][c], false, Bf[c],
                                                   (short)0, acc0, false, false);
      acc1 = __builtin_amdgcn_wmma_f32_16x16x4_f32(false, hA[1][c], false, Bf[c],
                                                   (short)0, acc1, false, false);
    }

    const int rowoff = (lane >> 4) * 8;
#pragma unroll
    for (int r = 0; r < 8; ++r) {
      const int ip0 = rowoff + r;       // i' = 0..15
      const int ip1 = 16 + rowoff + r;  // i' = 16..31
      yb[(size_t)(ip0 * 32 + jp) * D_FULL + mlo] = acc0[r];
      yb[(size_t)(ip1 * 32 + jp) * D_FULL + mlo] = acc1[r];
    }
  }
}

extern "C" void kernel_launch(void* const* d_in, const int* in_sizes, int n_in,
                              void* d_out, int out_size, void* d_ws, size_t ws_size,
                              hipStream_t stream) {
  const float* x = (const float*)d_in[0];
  float* y = (float*)d_out;
  const int B = in_sizes[0] / (N_FWHT * D_FULL);   // 128
  dim3 grid(D_FULL / DT, B);                       // (16, 128)
  fwht_wmma_kernel<<<grid, 256, 0, stream>>>(x, y);
}